// MGNN_32564442038707
// MI455X (gfx1250) — compile-verified
//
#include <hip/hip_runtime.h>
#include <hip/hip_bf16.h>
#include <math.h>

typedef __attribute__((ext_vector_type(2))) float v2f;
typedef __attribute__((ext_vector_type(8))) float v8f;

#define NN 8192      // nodes
#define NE 262144    // edges
#define NF 200       // input features
#define NH 128       // hidden
#define NO 16        // outputs
#define KC0 10       // level-0 clusters
#define KC1 5        // level-1 clusters

// ---------------------------------------------------------------------------
// fp32 WMMA: D(16x16) = A(16x4) * B(4x16) + C   (wave32, full fp32 precision)
// A frag: lane m (0-15) holds {A[m][k], A[m][k+1]} at koff=0, lanes 16-31 koff=2
// B frag: lane n holds {B[k][n], B[k+1][n]} = {W[n][k], W[n][k+1]} (same koff)
// ---------------------------------------------------------------------------
__device__ __forceinline__ v8f wmma4(v2f a, v2f b, v8f c) {
  return __builtin_amdgcn_wmma_f32_16x16x4_f32(false, a, false, b, (short)0, c,
                                               false, false);
}

__global__ void zero_f32(float* __restrict__ p, int n) {
  int stride = gridDim.x * blockDim.x;
  for (int i = blockIdx.x * blockDim.x + threadIdx.x; i < n; i += stride)
    p[i] = 0.0f;
}

// C[M,N] = A[M,K](ldA) * W[N,K](ldW)^T   — one wave per 16x16 tile
__global__ void gemm_wmma(const float* __restrict__ A, const float* __restrict__ W,
                          float* __restrict__ C, int M, int N, int K,
                          int ldA, int ldW, int ldC) {
  int wave = (blockIdx.x * blockDim.x + threadIdx.x) >> 5;
  int lane = threadIdx.x & 31;
  int ntn = N >> 4;
  int tiles = (M >> 4) * ntn;
  if (wave >= tiles) return;
  int tm = wave / ntn, tn = wave - tm * ntn;
  int mrow = (tm << 4) + (lane & 15);
  int ncol = (tn << 4) + (lane & 15);
  int koff = (lane >> 4) << 1;
  const float* Ap = A + (size_t)mrow * ldA + koff;
  const float* Wp = W + (size_t)ncol * ldW + koff;
  v8f acc = {0.f, 0.f, 0.f, 0.f, 0.f, 0.f, 0.f, 0.f};
  for (int k = 0; k < K; k += 4) {
    v2f a = *(const v2f*)(Ap + k);
    v2f b = *(const v2f*)(Wp + k);
    acc = wmma4(a, b, acc);
  }
  int rbase = (tm << 4) + ((lane >> 4) << 3);
  int col = (tn << 4) + (lane & 15);
  for (int r = 0; r < 8; ++r)
    C[(size_t)(rbase + r) * ldC + col] = acc[r];
}

// Out[dst[e]] += w[e] * Xin[src[e]]   (128 feats, wave-per-edge, float4/lane)
__global__ void edge_agg(const int* __restrict__ src, const int* __restrict__ dst,
                         const float* __restrict__ wt, const float* __restrict__ Xin,
                         float* __restrict__ Out) {
  int wave = (blockIdx.x * blockDim.x + threadIdx.x) >> 5;
  int lane = threadIdx.x & 31;
  int nw = (gridDim.x * blockDim.x) >> 5;
  for (int e = wave; e < NE; e += nw) {
    int s = src[e], d = dst[e];
    float w = wt[e];
    const float4 v = *(const float4*)(Xin + (size_t)s * NH + lane * 4);
    float* o = Out + (size_t)d * NH + lane * 4;
    unsafeAtomicAdd(o + 0, w * v.x);
    unsafeAtomicAdd(o + 1, w * v.y);
    unsafeAtomicAdd(o + 2, w * v.z);
    unsafeAtomicAdd(o + 3, w * v.w);
  }
}

// Out = relu(layernorm(X + bias) * g + b), wave-per-row (128 feats)
__global__ void ln_relu(const float* __restrict__ X, const float* __restrict__ bias,
                        const float* __restrict__ g, const float* __restrict__ b,
                        float* __restrict__ Out) {
  int wave = (blockIdx.x * blockDim.x + threadIdx.x) >> 5;
  int lane = threadIdx.x & 31;
  if (wave >= NN) return;
  const float* row = X + (size_t)wave * NH;
  float v[4];
  float s = 0.f;
  for (int j = 0; j < 4; ++j) { v[j] = row[lane * 4 + j] + bias[lane * 4 + j]; s += v[j]; }
  for (int m = 16; m >= 1; m >>= 1) s += __shfl_xor(s, m, 32);
  float mu = s * (1.0f / NH);
  float q = 0.f;
  for (int j = 0; j < 4; ++j) { float d = v[j] - mu; q += d * d; }
  for (int m = 16; m >= 1; m >>= 1) q += __shfl_xor(q, m, 32);
  float inv = 1.0f / sqrtf(q * (1.0f / NH) + 1e-5f);
  float* o = Out + (size_t)wave * NH;
  for (int j = 0; j < 4; ++j) {
    int f = lane * 4 + j;
    float y = (v[j] - mu) * inv * g[f] + b[f];
    o[f] = fmaxf(y, 0.f);
  }
}

// c0[i] = argmax_k(latent[i]·pW[k] + pb[k] + gum[i,k])  (softmax is monotone)
__global__ void pool0_argmax(const float* __restrict__ latent, const float* __restrict__ pW,
                             const float* __restrict__ pb, const float* __restrict__ gum,
                             int* __restrict__ c0) {
  int i = blockIdx.x * blockDim.x + threadIdx.x;
  if (i >= NN) return;
  const float* row = latent + (size_t)i * NH;
  float best = -3.0e38f;
  int bk = 0;
  for (int k = 0; k < KC0; ++k) {
    float d = pb[k] + gum[i * KC0 + k];
    const float* wr = pW + k * NH;
    for (int f = 0; f < NH; ++f) d += row[f] * wr[f];
    if (d > best) { best = d; bk = k; }
  }
  c0[i] = bk;
}

// latsum[c0[i]] += latent[i]  (LDS-staged; thread f exclusively owns column f)
__global__ void latsum0_kernel(const float* __restrict__ latent, const int* __restrict__ c0,
                               float* __restrict__ latsum) {
  __shared__ float acc[KC0 * NH];
  int f = threadIdx.x;  // 128 threads
  for (int a = 0; a < KC0; ++a) acc[a * NH + f] = 0.f;
  int per = NN / gridDim.x;
  int start = blockIdx.x * per;
  for (int i = start; i < start + per; ++i)
    acc[c0[i] * NH + f] += latent[(size_t)i * NH + f];
  __syncthreads();
  for (int a = 0; a < KC0; ++a)
    unsafeAtomicAdd(&latsum[a * NH + f], acc[a * NH + f]);
}

// adj0[c0[src],c0[dst]] += w   (LDS atomic staging -> 100 global atomics)
__global__ void adj0_kernel(const int* __restrict__ src, const int* __restrict__ dst,
                            const float* __restrict__ wt, const int* __restrict__ c0,
                            float* __restrict__ adj0) {
  __shared__ float sadj[KC0 * KC0];
  int t = threadIdx.x;
  if (t < KC0 * KC0) sadj[t] = 0.f;
  __syncthreads();
  int stride = gridDim.x * blockDim.x;
  for (int e = blockIdx.x * blockDim.x + t; e < NE; e += stride)
    atomicAdd(&sadj[c0[src[e]] * KC0 + c0[dst[e]]], wt[e]);
  __syncthreads();
  if (t < KC0 * KC0) unsafeAtomicAdd(&adj0[t], sadj[t]);
}

// Both pooling levels (tiny): normalize, encoders, level-1 argmax, fc partials.
// Single block of 128 threads (thread f owns feature column f).
__global__ void levels_kernel(const float* __restrict__ latsum0, const float* __restrict__ adj0g,
                              const float* __restrict__ p1W, const float* __restrict__ p1b,
                              const float* __restrict__ gum1,
                              const float* __restrict__ e0W, const float* __restrict__ e0b,
                              const float* __restrict__ e1W, const float* __restrict__ e1b,
                              const float* __restrict__ fcW,
                              float* __restrict__ t0g, float* __restrict__ t1g,
                              int* __restrict__ c1g) {
  __shared__ float xc0[KC0 * NH], tmp0[KC0 * NH], lat0[KC0 * NH];
  __shared__ float xc1[KC1 * NH], tmp1[KC1 * NH], lat1[KC1 * NH];
  __shared__ float sadj[KC0 * KC0], sadj1[KC1 * KC1];
  __shared__ float red[NH];
  __shared__ int c1[KC0];
  int f = threadIdx.x;

  // xc0 = row_normalize(latsum0)
  for (int a = 0; a < KC0; ++a) {
    float v = latsum0[a * NH + f];
    red[f] = v * v; __syncthreads();
    for (int st = 64; st >= 1; st >>= 1) { if (f < st) red[f] += red[f + st]; __syncthreads(); }
    float n = sqrtf(red[0]); __syncthreads();
    xc0[a * NH + f] = v / fmaxf(n, 1e-12f);
  }

  // sadj = adj0 / (sum(adj0) + 100*1e-8)
  red[f] = (f < KC0 * KC0) ? adj0g[f] : 0.f; __syncthreads();
  for (int st = 64; st >= 1; st >>= 1) { if (f < st) red[f] += red[f + st]; __syncthreads(); }
  float denom0 = red[0] + 1e-8f * (KC0 * KC0); __syncthreads();
  if (f < KC0 * KC0) sadj[f] = adj0g[f] / denom0;
  __syncthreads();

  // lat0 = relu((sadj @ xc0) @ e0W^T + e0b)
  for (int a = 0; a < KC0; ++a) {
    float s = 0.f;
    for (int b = 0; b < KC0; ++b) s += sadj[a * KC0 + b] * xc0[b * NH + f];
    tmp0[a * NH + f] = s;
  }
  __syncthreads();
  for (int a = 0; a < KC0; ++a) {
    float s = e0b[f];
    const float* wr = e0W + f * NH;
    for (int g = 0; g < NH; ++g) s += tmp0[a * NH + g] * wr[g];
    lat0[a * NH + f] = fmaxf(s, 0.f);
  }
  __syncthreads();

  // level-1 assignment: c1[a] = argmax_k(lat0[a]·p1W[k] + p1b[k] + gum1[a,k])
  if (f < KC0) {
    float best = -3.0e38f; int bk = 0;
    for (int k = 0; k < KC1; ++k) {
      float d = p1b[k] + gum1[f * KC1 + k];
      const float* wr = p1W + k * NH;
      for (int g = 0; g < NH; ++g) d += lat0[f * NH + g] * wr[g];
      if (d > best) { best = d; bk = k; }
    }
    c1[f] = bk;
    c1g[f] = bk;
  }
  __syncthreads();

  // xc1 = row_normalize(assign1^T @ lat0)
  float acc1[KC1];
  for (int p = 0; p < KC1; ++p) acc1[p] = 0.f;
  for (int a = 0; a < KC0; ++a) acc1[c1[a]] += lat0[a * NH + f];
  for (int p = 0; p < KC1; ++p) {
    float v = acc1[p];
    red[f] = v * v; __syncthreads();
    for (int st = 64; st >= 1; st >>= 1) { if (f < st) red[f] += red[f + st]; __syncthreads(); }
    float n = sqrtf(red[0]); __syncthreads();
    xc1[p * NH + f] = v / fmaxf(n, 1e-12f);
  }

  // sadj1 = pool(sadj) / (sum + 25*1e-8)
  if (f < KC1 * KC1) sadj1[f] = 0.f;
  __syncthreads();
  if (f < KC0 * KC0) {
    int a = f / KC0, b = f - a * KC0;
    atomicAdd(&sadj1[c1[a] * KC1 + c1[b]], sadj[f]);
  }
  __syncthreads();
  red[f] = (f < KC1 * KC1) ? sadj1[f] : 0.f; __syncthreads();
  for (int st = 64; st >= 1; st >>= 1) { if (f < st) red[f] += red[f + st]; __syncthreads(); }
  float denom1 = red[0] + 1e-8f * (KC1 * KC1); __syncthreads();
  if (f < KC1 * KC1) sadj1[f] /= denom1;
  __syncthreads();

  // lat1 = relu((sadj1 @ xc1) @ e1W^T + e1b)
  for (int p = 0; p < KC1; ++p) {
    float s = 0.f;
    for (int q = 0; q < KC1; ++q) s += sadj1[p * KC1 + q] * xc1[q * NH + f];
    tmp1[p * NH + f] = s;
  }
  __syncthreads();
  for (int p = 0; p < KC1; ++p) {
    float s = e1b[f];
    const float* wr = e1W + f * NH;
    for (int g = 0; g < NH; ++g) s += tmp1[p * NH + g] * wr[g];
    lat1[p * NH + f] = fmaxf(s, 0.f);
  }
  __syncthreads();

  // fc partials for broadcast latents: t0[a,o] = lat0[a]·fcW[o,128:256], t1 likewise
  for (int t = f; t < KC0 * NO; t += NH) {
    int a = t / NO, o = t - a * NO;
    float s = 0.f;
    const float* wr = fcW + o * (3 * NH) + NH;
    for (int g = 0; g < NH; ++g) s += lat0[a * NH + g] * wr[g];
    t0g[t] = s;
  }
  for (int t = f; t < KC1 * NO; t += NH) {
    int p = t / NO, o = t - p * NO;
    float s = 0.f;
    const float* wr = fcW + o * (3 * NH) + 2 * NH;
    for (int g = 0; g < NH; ++g) s += lat1[p * NH + g] * wr[g];
    t1g[t] = s;
  }
}

// out[i] = latent[i] @ fcW[:, :128]^T + t0[c0[i]] + t1[c1[c0[i]]] + fcb
// WMMA over K=128, N=16 (one tile), fused gather epilogue.
__global__ void fc_wmma(const float* __restrict__ latent, const float* __restrict__ fcW,
                        const float* __restrict__ fcb, const float* __restrict__ t0,
                        const float* __restrict__ t1, const int* __restrict__ c0,
                        const int* __restrict__ c1, float* __restrict__ out) {
  int wave = (blockIdx.x * blockDim.x + threadIdx.x) >> 5;  // = tileM
  int lane = threadIdx.x & 31;
  if (wave >= NN / 16) return;
  int mrow = wave * 16 + (lane & 15);
  int koff = (lane >> 4) << 1;
  const float* Ap = latent + (size_t)mrow * NH + koff;
  const float* Wp = fcW + (size_t)(lane & 15) * (3 * NH) + koff;
  v8f acc = {0.f, 0.f, 0.f, 0.f, 0.f, 0.f, 0.f, 0.f};
  for (int k = 0; k < NH; k += 4) {
    v2f a = *(const v2f*)(Ap + k);
    v2f b = *(const v2f*)(Wp + k);
    acc = wmma4(a, b, acc);
  }
  int rbase = wave * 16 + ((lane >> 4) << 3);
  int col = lane & 15;
  float bias = fcb[col];
  for (int r = 0; r < 8; ++r) {
    int row = rbase + r;
    int a0 = c0[row];
    int p1 = c1[a0];
    out[(size_t)row * NO + col] = acc[r] + t0[a0 * NO + col] + t1[p1 * NO + col] + bias;
  }
}

extern "C" void kernel_launch(void* const* d_in, const int* in_sizes, int n_in,
                              void* d_out, int out_size, void* d_ws, size_t ws_size,
                              hipStream_t stream) {
  const float* x   = (const float*)d_in[0];
  const int*   ei  = (const int*)d_in[1];
  const float* ew  = (const float*)d_in[2];
  const float* g0  = (const float*)d_in[3];
  const float* g1  = (const float*)d_in[4];
  const float* Wg1 = (const float*)d_in[5];
  const float* bg1 = (const float*)d_in[6];
  const float* l1g = (const float*)d_in[7];
  const float* l1b = (const float*)d_in[8];
  const float* Wg2 = (const float*)d_in[9];
  const float* bg2 = (const float*)d_in[10];
  const float* l2g = (const float*)d_in[11];
  const float* l2b = (const float*)d_in[12];
  const float* p0W = (const float*)d_in[13];
  const float* p0b = (const float*)d_in[14];
  const float* p1W = (const float*)d_in[15];
  const float* p1b = (const float*)d_in[16];
  const float* e0W = (const float*)d_in[17];
  const float* e0b = (const float*)d_in[18];
  const float* e1W = (const float*)d_in[19];
  const float* e1b = (const float*)d_in[20];
  const float* fcW = (const float*)d_in[21];
  const float* fcb = (const float*)d_in[22];
  float* out = (float*)d_out;
  float* ws  = (float*)d_ws;

  // workspace layout (floats) — ~25.2 MB total
  float* xw1     = ws;                    // 8192*128
  float* h1      = ws + 1048576;          // 8192*128
  float* hw2     = ws + 2097152;          // 8192*128
  float* latent  = ws + 3145728;          // 8192*128
  float* agg1    = ws + 4194304;          // 8192*128 (zeroed)
  float* agg2    = ws + 5242880;          // 8192*128 (zeroed)
  float* latsum0 = ws + 6291456;          // 10*128   (zeroed)
  float* adj0    = latsum0 + KC0 * NH;    // 100      (zeroed)
  float* t0      = adj0 + KC0 * KC0;      // 160
  float* t1      = t0 + KC0 * NO;         // 80
  int*   c0      = (int*)(t1 + KC1 * NO); // 8192 ints
  int*   c1      = c0 + NN;               // 10 ints

  const int* esrc = ei;
  const int* edst = ei + NE;

  int zn = 2 * 1048576 + KC0 * NH + KC0 * KC0;  // agg1,agg2,latsum0,adj0 contiguous
  zero_f32<<<2048, 256, 0, stream>>>(agg1, zn);

  // layer 1: (x @ Wg1^T) then aggregate, then LN(+bg1)+relu
  gemm_wmma<<<512, 256, 0, stream>>>(x, Wg1, xw1, NN, NH, NF, NF, NF, NH);
  edge_agg<<<2048, 256, 0, stream>>>(esrc, edst, ew, xw1, agg1);
  ln_relu<<<1024, 256, 0, stream>>>(agg1, bg1, l1g, l1b, h1);

  // layer 2
  gemm_wmma<<<512, 256, 0, stream>>>(h1, Wg2, hw2, NN, NH, NH, NH, NH, NH);
  edge_agg<<<2048, 256, 0, stream>>>(esrc, edst, ew, hw2, agg2);
  ln_relu<<<1024, 256, 0, stream>>>(agg2, bg2, l2g, l2b, latent);

  // pooling hierarchy
  pool0_argmax<<<32, 256, 0, stream>>>(latent, p0W, p0b, g0, c0);
  latsum0_kernel<<<128, 128, 0, stream>>>(latent, c0, latsum0);
  adj0_kernel<<<256, 256, 0, stream>>>(esrc, edst, ew, c0, adj0);
  levels_kernel<<<1, 128, 0, stream>>>(latsum0, adj0, p1W, p1b, g1,
                                       e0W, e0b, e1W, e1b, fcW, t0, t1, c1);

  // final projection with fused broadcast-latent gather
  fc_wmma<<<64, 256, 0, stream>>>(latent, fcW, fcb, t0, t1, c0, c1, out);
}